// LGR_55989193671419
// MI455X (gfx1250) — compile-verified
//
#include <hip/hip_runtime.h>
#include <math.h>

// ---------------------------------------------------------------------------
// LGR essential-matrix kernel for MI455X (gfx1250, wave32).
//   Phase 1: streaming accumulation of XwX = (wX)^T X via V_WMMA_F32_16X16X4_F32
//            (full f32 precision; kernel is HBM-bound: 128 MB -> ~5.5 us floor)
//   Phase 2: per-batch 9x9 Jacobi eigensolve, smallest eigenvector, normalize
// ---------------------------------------------------------------------------

typedef __attribute__((ext_vector_type(2))) float v2f;
typedef __attribute__((ext_vector_type(8))) float v8f;

#define BATCH    64
#define NPTS     100000
#define WG_PER_B 32
#define CHUNK    ((NPTS + WG_PER_B - 1) / WG_PER_B)   /* 3125 */
#define WAVES    8
#define BLK      (WAVES * 32)

__global__ void lgr_zero(float* __restrict__ acc, int n) {
  int i = blockIdx.x * blockDim.x + threadIdx.x;
  if (i < n) acc[i] = 0.0f;
}

__global__ __launch_bounds__(BLK)
void lgr_accum(const float4* __restrict__ xin,
               const float*  __restrict__ logits,
               float*        __restrict__ acc /* BATCH x 81 */) {
  // Per-wave private staging: 32 points x 20 floats (80 B, 16-B aligned rows):
  //   [0..7] variable features (two float4 stores), [8] = 1.0 constant,
  //   [9..15] zero pad (only feeds C rows/cols >= 9, never extracted),
  //   [16] = weight.  Stride-20 dwords => conflict-light on 64 banks.
  __shared__ __align__(16) float ldsX[WAVES][32][20];
  __shared__ float red[WAVES][81];

  const int b    = blockIdx.x / WG_PER_B;
  const int s    = blockIdx.x % WG_PER_B;
  const int wave = threadIdx.x >> 5;
  const int lane = threadIdx.x & 31;
  const int half = lane >> 4;    // selects K pair {0,1} vs {2,3} in the frag
  const int fl   = lane & 15;    // feature / column index within tile

  const int pBeg = s * CHUNK;
  const int pEnd = (pBeg + CHUNK < NPTS) ? (pBeg + CHUNK) : NPTS;

  // One-time constant slots (wave-private, no barrier needed).
  ldsX[wave][lane][8] = 1.0f;            // X[8] == 1 for every point
  #pragma unroll
  for (int i = 9; i < 16; ++i) ldsX[wave][lane][i] = 0.0f;

  v8f c = {};  // 16x16 f32 accumulator tile; we extract the 9x9 corner
  const size_t bOfs = (size_t)b * NPTS;

  for (int base = pBeg; base < pEnd; base += BLK) {
    const int p  = base + wave * 32 + lane;
    const int pc = (p < pEnd) ? p : (pEnd - 1);   // clamped, always valid

    const float4 v  = xin[bOfs + pc];
    const float  lg = logits[bOfs + pc];

    // Branch-free relu(tanh(lg)): for x<=0 result is 0, so clamp to [0,15]
    // and evaluate tanh(x) = (1 - e^-2x) / (1 + e^-2x).
    const float xc = fminf(fmaxf(lg, 0.0f), 15.0f);
    const float ex = __expf(-2.0f * xc);
    float w = (1.0f - ex) * __builtin_amdgcn_rcpf(1.0f + ex);
    // Out-of-range point: zero weight kills its entire rank-1 contribution
    // (A-fragment = w*X), so clamped/garbage features are harmless.
    w = (p < pEnd) ? w : 0.0f;

    const float x1 = v.x, y1 = v.y, x2 = v.z, y2 = v.w;
    float4 f0; f0.x = x2 * x1; f0.y = x2 * y1; f0.z = x2; f0.w = y2 * x1;
    float4 f1; f1.x = y2 * y1; f1.y = y2;      f1.z = x1; f1.w = y1;

    float4* row = (float4*)&ldsX[wave][lane][0];
    row[0] = f0;
    row[1] = f1;
    ldsX[wave][lane][16] = w;
    // No barrier: producer and consumer are the same wave; hardware keeps
    // per-wave DS ops in order and the compiler cannot reorder may-aliasing
    // LDS accesses. Waves stay fully decoupled for latency hiding.

    // 32 staged points -> 8 WMMAs of K=4 each.
    // A (16x4 f32): lane holds M=fl, K = {half*2, half*2+1} in v[0],v[1]
    // B (4x16 f32): lane holds N=fl, same K pair mapping.
    #pragma unroll
    for (int kk = 0; kk < 8; ++kk) {
      const int k0 = kk * 4 + half * 2;
      const int k1 = k0 + 1;
      const float x0  = ldsX[wave][k0][fl];
      const float x1v = ldsX[wave][k1][fl];
      const float w0  = ldsX[wave][k0][16];
      const float w1  = ldsX[wave][k1][16];
      v2f a;  a[0]  = w0 * x0; a[1]  = w1 * x1v;   // (wX)^T fragment
      v2f bv; bv[0] = x0;      bv[1] = x1v;        // X fragment
      c = __builtin_amdgcn_wmma_f32_16x16x4_f32(
              /*neg_a=*/false, a, /*neg_b=*/false, bv,
              /*c_mod=*/(short)0, c, /*reuse_a=*/false, /*reuse_b=*/false);
    }
  }

  // Scatter 9x9 corner of the C tile (VGPR r: lanes 0-15 -> row r, lanes 16-31 -> row r+8)
  if (half == 0) {
    if (fl < 9) {
      #pragma unroll
      for (int r = 0; r < 8; ++r) red[wave][r * 9 + fl] = c[r];
    }
  } else {
    if (fl < 9) red[wave][72 + fl] = c[0];        // row 8
  }
  __syncthreads();

  // Cross-wave reduce, then one atomic per element per workgroup.
  for (int t = threadIdx.x; t < 81; t += BLK) {
    float sum = 0.0f;
    #pragma unroll
    for (int wv = 0; wv < WAVES; ++wv) sum += red[wv][t];
    atomicAdd(&acc[b * 81 + t], sum);
  }
}

// ---------------------------------------------------------------------------
// Phase 2: cyclic Jacobi on 9x9 symmetric, one thread per batch (trivial work).
// ---------------------------------------------------------------------------
__global__ void lgr_eig(const float* __restrict__ acc, float* __restrict__ out) {
  const int b = blockIdx.x * blockDim.x + threadIdx.x;
  if (b >= BATCH) return;

  float A[9][9], V[9][9];
  for (int i = 0; i < 9; ++i)
    for (int j = 0; j < 9; ++j) {
      A[i][j] = acc[b * 81 + i * 9 + j];
      V[i][j] = (i == j) ? 1.0f : 0.0f;
    }

  for (int sweep = 0; sweep < 12; ++sweep) {
    for (int p = 0; p < 8; ++p) {
      for (int q = p + 1; q < 9; ++q) {
        const float apq = A[p][q];
        const float scale = fabsf(A[p][p]) + fabsf(A[q][q]) + 1e-30f;
        if (fabsf(apq) < 1e-12f * scale) continue;
        const float tau = (A[q][q] - A[p][p]) / (2.0f * apq);
        const float rt  = sqrtf(1.0f + tau * tau);
        const float t   = (tau >= 0.0f) ? 1.0f / (tau + rt) : 1.0f / (tau - rt);
        const float cc  = 1.0f / sqrtf(1.0f + t * t);
        const float ss  = t * cc;
        // A <- J^T A J  (column then row update)
        for (int i = 0; i < 9; ++i) {
          const float aip = A[i][p], aiq = A[i][q];
          A[i][p] = cc * aip - ss * aiq;
          A[i][q] = ss * aip + cc * aiq;
        }
        for (int i = 0; i < 9; ++i) {
          const float api = A[p][i], aqi = A[q][i];
          A[p][i] = cc * api - ss * aqi;
          A[q][i] = ss * api + cc * aqi;
        }
        // V <- V J
        for (int i = 0; i < 9; ++i) {
          const float vip = V[i][p], viq = V[i][q];
          V[i][p] = cc * vip - ss * viq;
          V[i][q] = ss * vip + cc * viq;
        }
      }
    }
  }

  // Smallest eigenvalue's eigenvector, normalized.
  int m = 0;
  float mv = A[0][0];
  for (int i = 1; i < 9; ++i)
    if (A[i][i] < mv) { mv = A[i][i]; m = i; }

  float nrm = 0.0f;
  float e[9];
  for (int i = 0; i < 9; ++i) { e[i] = V[i][m]; nrm += e[i] * e[i]; }
  const float inv = 1.0f / sqrtf(nrm + 1e-30f);
  for (int i = 0; i < 9; ++i) out[b * 9 + i] = e[i] * inv;
}

extern "C" void kernel_launch(void* const* d_in, const int* in_sizes, int n_in,
                              void* d_out, int out_size, void* d_ws, size_t ws_size,
                              hipStream_t stream) {
  const float4* xin    = (const float4*)d_in[0];
  const float*  logits = (const float*)d_in[1];
  float*        out    = (float*)d_out;
  float*        acc    = (float*)d_ws;  // BATCH * 81 floats

  const int nAcc = BATCH * 81;
  lgr_zero<<<(nAcc + 255) / 256, 256, 0, stream>>>(acc, nAcc);
  lgr_accum<<<BATCH * WG_PER_B, BLK, 0, stream>>>(xin, logits, acc);
  lgr_eig<<<(BATCH + 31) / 32, 32, 0, stream>>>(acc, out);
}